// CoreDiffusion_21414706937939
// MI455X (gfx1250) — compile-verified
//
#include <hip/hip_runtime.h>
#include <math.h>

// Problem constants (match reference): B=2, C=4, N=4096, D=64, H=64
#define NB 2
#define NC 4
#define NN 4096
#define ND 64
#define NH 64

typedef __attribute__((ext_vector_type(2))) float v2f;
typedef __attribute__((ext_vector_type(8))) float v8f;
typedef int v4i __attribute__((vector_size(16)));   // 16B payload type for b128

// gfx1250 async global->LDS path (ASYNCcnt-tracked), with sync fallback.
#if defined(__has_builtin)
#  if __has_builtin(__builtin_amdgcn_global_load_async_to_lds_b128) && \
      __has_builtin(__builtin_amdgcn_s_wait_asynccnt)
#    define USE_ASYNC 1
#  endif
#endif
#ifndef USE_ASYNC
#  define USE_ASYNC 0
#endif

static __device__ __forceinline__ v8f wmma_f32(v2f a, v2f b, v8f c) {
  // D = A(16x4 f32) * B(4x16 f32) + C(16x16 f32)
  return __builtin_amdgcn_wmma_f32_16x16x4_f32(
      /*neg_a=*/false, a, /*neg_b=*/false, b,
      /*c_mod=*/(short)0, c, /*reuse_a=*/false, /*reuse_b=*/false);
}

static __device__ __forceinline__ v8f splat8(float s) {
  v8f v;
#pragma unroll
  for (int i = 0; i < 8; ++i) v[i] = s;
  return v;
}

#if USE_ASYNC
static __device__ __forceinline__ void async_b128(const float* g, float* l) {
  __builtin_amdgcn_global_load_async_to_lds_b128(
      (__attribute__((address_space(1))) v4i*)g,
      (__attribute__((address_space(3))) v4i*)l,
      /*offset=*/0, /*cpol=*/0);
}
#endif

// ---------------------------------------------------------------------------
// Kernel 1: msg[b,c] = adj[b,c] (4096x4096) @ x[b] (4096x64), fp32 WMMA.
// Block = 256 threads (8 waves); 128 output rows x 64 cols per block.
// adj tiles: double-buffered GLOBAL_LOAD_ASYNC_TO_LDS_B128 (no VGPR staging).
// x tiles: VGPR path, stored TRANSPOSED so B fragments are contiguous b64.
// ---------------------------------------------------------------------------
__global__ __launch_bounds__(256) void k_diffusion_gemm(
    const float* __restrict__ adj, const float* __restrict__ x,
    float* __restrict__ msg) {
  const int tid  = threadIdx.x;
  const int lane = tid & 31;
  const int wave = tid >> 5;
  const int half = lane >> 4;      // lane-half selects K sub-pair / row-half
  const int l16  = lane & 15;
  const int koff = half * 2;

  const int rowBase = blockIdx.x * 128;
  const int c = blockIdx.y;
  const int b = blockIdx.z;

  // Row pad 36: keeps b128 LDS dests 16B-aligned (36*4=144=9*16) and makes
  // the two lane-halves of every fragment read hit disjoint bank sets.
  __shared__ float As[2][128][36];   // adj tile: [row][k]
  __shared__ float XsT[2][64][36];   // x tile TRANSPOSED: [col][k]

  const float* __restrict__ adjB =
      adj + (((size_t)(b * NC + c)) * NN + rowBase) * NN;
  const float* __restrict__ xB = x + (size_t)b * NN * ND;

  auto stage_A = [&](int k0, int buf) {
#pragma unroll
    for (int i = 0; i < 4; ++i) {           // 128x32 = 1024 float4, 4/thread
      int idx = tid + 256 * i;
      int r   = idx >> 3;
      int c4  = (idx & 7) << 2;
#if USE_ASYNC
      async_b128(adjB + (size_t)r * NN + (k0 + c4), &As[buf][r][c4]);
#else
      float4 v = *reinterpret_cast<const float4*>(adjB + (size_t)r * NN + (k0 + c4));
      As[buf][r][c4 + 0] = v.x; As[buf][r][c4 + 1] = v.y;
      As[buf][r][c4 + 2] = v.z; As[buf][r][c4 + 3] = v.w;
#endif
    }
  };
  auto stage_X = [&](int k0, int buf) {
#pragma unroll
    for (int i = 0; i < 2; ++i) {           // 32x64 = 512 float4, 2/thread
      int idx = tid + 256 * i;
      int r   = idx >> 4;
      int c4  = (idx & 15) << 2;
      float4 v = *reinterpret_cast<const float4*>(xB + (size_t)(k0 + r) * ND + c4);
      XsT[buf][c4 + 0][r] = v.x;            // transpose on the way in
      XsT[buf][c4 + 1][r] = v.y;
      XsT[buf][c4 + 2][r] = v.z;
      XsT[buf][c4 + 3][r] = v.w;
    }
  };

  v8f acc[4];
#pragma unroll
  for (int n = 0; n < 4; ++n) acc[n] = splat8(0.f);

  // Prologue: stage chunk 0 into buffer 0.
  stage_A(0, 0);
  stage_X(0, 0);

  for (int k0 = 0; k0 < NN; k0 += 32) {
    const int  cur  = (k0 >> 5) & 1;
    const bool more = (k0 + 32) < NN;
    if (more) {                      // prefetch next chunk into other buffer
      stage_A(k0 + 32, cur ^ 1);
      stage_X(k0 + 32, cur ^ 1);
    }
#if USE_ASYNC
    // Async loads complete in order: <=4 outstanding => current chunk landed.
    if (more) __builtin_amdgcn_s_wait_asynccnt(4);
    else      __builtin_amdgcn_s_wait_asynccnt(0);
#endif
    __syncthreads();

    const int arow = (wave << 4) + l16;     // A row = lane%16 within wave tile
#pragma unroll
    for (int kk = 0; kk < 32; kk += 4) {
      v2f a;                                 // contiguous -> ds_load b64
      a[0] = As[cur][arow][kk + koff];
      a[1] = As[cur][arow][kk + koff + 1];
#pragma unroll
      for (int n = 0; n < 4; ++n) {
        v2f bf;                              // contiguous -> ds_load b64
        bf[0] = XsT[cur][(n << 4) + l16][kk + koff];
        bf[1] = XsT[cur][(n << 4) + l16][kk + koff + 1];
        acc[n] = wmma_f32(a, bf, acc[n]);
      }
    }
    __syncthreads();
  }

  // D layout: vgpr r -> row r + half*8; col = 16n + lane%16.
  float* __restrict__ dst =
      msg + (((size_t)(b * NC + c)) * NN + rowBase + (wave << 4) + half * 8) * ND + l16;
#pragma unroll
  for (int n = 0; n < 4; ++n)
#pragma unroll
    for (int r = 0; r < 8; ++r)
      dst[(size_t)r * ND + (n << 4)] = acc[n][r];
}

// ---------------------------------------------------------------------------
// Kernel 2: cumsum over cores + ReLU + 4-step GRU (WMMA gates) + LayerNorm.
// Block = 128 threads (4 waves); one wave per 16 (b,node) rows.
// ---------------------------------------------------------------------------
__global__ __launch_bounds__(128) void k_gru_ln(
    const float* __restrict__ msg,
    const float* __restrict__ w_ih, const float* __restrict__ w_hh,
    const float* __restrict__ b_ih, const float* __restrict__ b_hh,
    const float* __restrict__ gamma, const float* __restrict__ beta,
    float* __restrict__ out) {
  const int tid  = threadIdx.x;
  const int lane = tid & 31;
  const int wave = tid >> 5;
  const int half = lane >> 4;
  const int l16  = lane & 15;
  const int koff = half * 2;

  const int flatBase = blockIdx.x * 64 + (wave << 4);  // over B*N
  const int b    = flatBase >> 12;                     // /4096
  const int node = flatBase & (NN - 1);

  __shared__ float Cum[4][16][64];   // running cumsum (plain layout)
  __shared__ float Hx[4][16][68];    // relu(cumsum) = GRU input (A-frag src)
  __shared__ float Hp[4][16][68];    // previous hidden state (A-frag src)
  __shared__ float Mu[4][16];
  __shared__ float Rs[4][16];

#pragma unroll
  for (int j = 0; j < 32; ++j) {
    int pos = lane + (j << 5);
    int row = pos >> 6, col = pos & 63;
    Cum[wave][row][col] = 0.f;
    Hp[wave][row][col]  = 0.f;
  }

  v8f outsum[4], hreg[4];
#pragma unroll
  for (int t = 0; t < 4; ++t) { outsum[t] = splat8(0.f); hreg[t] = splat8(0.f); }

  __syncthreads();

  for (int c = 0; c < NC; ++c) {
    const float* __restrict__ m =
        msg + (((size_t)(b * NC + c)) * NN + node) * ND;
#pragma unroll
    for (int j = 0; j < 32; ++j) {
      int pos = lane + (j << 5);
      int row = pos >> 6, col = pos & 63;
      float v = Cum[wave][row][col] + m[(size_t)row * ND + col];
      Cum[wave][row][col] = v;
      Hx[wave][row][col]  = fmaxf(v, 0.f);
    }
    __syncthreads();

    v8f nh[4];
#pragma unroll
    for (int t = 0; t < 4; ++t) {
      const int nr_r = (t << 4) + l16;   // w rows for reset gate cols
      const int nr_z = 64 + nr_r;        // update gate
      const int nr_n = 128 + nr_r;       // new gate
      // biases ride in as the WMMA C operand (splat per output column)
      v8f ir = splat8(b_ih[nr_r]), hr = splat8(b_hh[nr_r]);
      v8f iz = splat8(b_ih[nr_z]), hz = splat8(b_hh[nr_z]);
      v8f in_ = splat8(b_ih[nr_n]), hn = splat8(b_hh[nr_n]);

#pragma unroll
      for (int k = 0; k < NH; k += 4) {
        v2f a, ap;
        a[0]  = Hx[wave][l16][k + koff];  a[1]  = Hx[wave][l16][k + koff + 1];
        ap[0] = Hp[wave][l16][k + koff];  ap[1] = Hp[wave][l16][k + koff + 1];
        v2f w;
        // B[k][n] = W[n][k]  (gi = hx @ W_ih^T)
        w[0] = w_ih[(size_t)nr_r * NH + k + koff];
        w[1] = w_ih[(size_t)nr_r * NH + k + koff + 1];
        ir = wmma_f32(a, w, ir);
        w[0] = w_hh[(size_t)nr_r * NH + k + koff];
        w[1] = w_hh[(size_t)nr_r * NH + k + koff + 1];
        hr = wmma_f32(ap, w, hr);
        w[0] = w_ih[(size_t)nr_z * NH + k + koff];
        w[1] = w_ih[(size_t)nr_z * NH + k + koff + 1];
        iz = wmma_f32(a, w, iz);
        w[0] = w_hh[(size_t)nr_z * NH + k + koff];
        w[1] = w_hh[(size_t)nr_z * NH + k + koff + 1];
        hz = wmma_f32(ap, w, hz);
        w[0] = w_ih[(size_t)nr_n * NH + k + koff];
        w[1] = w_ih[(size_t)nr_n * NH + k + koff + 1];
        in_ = wmma_f32(a, w, in_);
        w[0] = w_hh[(size_t)nr_n * NH + k + koff];
        w[1] = w_hh[(size_t)nr_n * NH + k + koff + 1];
        hn = wmma_f32(ap, w, hn);
      }

      // elementwise gate math (gi/gh/h all share the D-matrix layout)
      v8f hv;
#pragma unroll
      for (int e = 0; e < 8; ++e) {
        float rr = 1.f / (1.f + __expf(-(ir[e] + hr[e])));
        float zz = 1.f / (1.f + __expf(-(iz[e] + hz[e])));
        float nn = tanhf(in_[e] + rr * hn[e]);
        hv[e] = (1.f - zz) * nn + zz * hreg[t][e];
      }
      outsum[t] += hv;
      nh[t] = hv;
    }

    // commit new hidden state (register copy + LDS copy for next A-frags)
#pragma unroll
    for (int t = 0; t < 4; ++t) {
      hreg[t] = nh[t];
#pragma unroll
      for (int r = 0; r < 8; ++r)
        Hp[wave][r + half * 8][(t << 4) + l16] = nh[t][r];
    }
    __syncthreads();
  }

  // LayerNorm over H=64 per node; stash sum-of-steps in Hx (plain layout)
#pragma unroll
  for (int t = 0; t < 4; ++t)
#pragma unroll
    for (int r = 0; r < 8; ++r)
      Hx[wave][r + half * 8][(t << 4) + l16] = outsum[t][r];
  __syncthreads();

  if (lane < 16) {
    float s = 0.f, s2 = 0.f;
#pragma unroll
    for (int col = 0; col < 64; ++col) {
      float v = Hx[wave][lane][col];
      s += v; s2 += v * v;
    }
    float mu  = s * (1.f / 64.f);
    float var = s2 * (1.f / 64.f) - mu * mu;
    Mu[wave][lane] = mu;
    Rs[wave][lane] = rsqrtf(var + 1e-5f);
  }
  __syncthreads();

#pragma unroll
  for (int j = 0; j < 32; ++j) {
    int pos = lane + (j << 5);
    int row = pos >> 6, col = pos & 63;
    float v = (Hx[wave][row][col] - Mu[wave][row]) * Rs[wave][row];
    out[((size_t)(flatBase + row)) * NH + col] = v * gamma[col] + beta[col];
  }
}

// ---------------------------------------------------------------------------
extern "C" void kernel_launch(void* const* d_in, const int* in_sizes, int n_in,
                              void* d_out, int out_size, void* d_ws, size_t ws_size,
                              hipStream_t stream) {
  (void)in_sizes; (void)n_in; (void)out_size; (void)ws_size;
  const float* x     = (const float*)d_in[0];
  const float* adj   = (const float*)d_in[1];
  const float* w_ih  = (const float*)d_in[2];
  const float* w_hh  = (const float*)d_in[3];
  const float* b_ih  = (const float*)d_in[4];
  const float* b_hh  = (const float*)d_in[5];
  const float* gamma = (const float*)d_in[6];
  const float* beta  = (const float*)d_in[7];
  float* out = (float*)d_out;
  float* msg = (float*)d_ws;  // B*C*N*D fp32 = 8 MB scratch

  dim3 g1(NN / 128, NC, NB);            // 32 x 4 x 2 = 256 workgroups
  k_diffusion_gemm<<<g1, 256, 0, stream>>>(adj, x, msg);

  dim3 g2((NB * NN) / 64);              // 128 workgroups
  k_gru_ln<<<g2, 128, 0, stream>>>(msg, w_ih, w_hh, b_ih, b_hh, gamma, beta, out);
}